// TSP_Decoder_62732292325623
// MI455X (gfx1250) — compile-verified
//
#include <hip/hip_runtime.h>
#include <hip/hip_bf16.h>
#include <math.h>

typedef __attribute__((ext_vector_type(2))) float v2f;
typedef __attribute__((ext_vector_type(8))) float v8f;

#define BB      64
#define PP      100
#define NN      1000
#define DD      128
#define HH      8
#define QDD     16
#define NTOTAL  1196
#define PT      7            // ceil(100/16)
#define PPAD    112
#define NT_K    75           // ceil(1196/16)
#define NT_N    63           // ceil(1000/16)
#define SROW    1204         // LDS row stride for attention scores (bank-safe)
#define SCROW   1012         // LDS row stride for pointer scores
#define INV_SQRT_D 0.08838834764831845f  // 1/sqrt(128)
#define CLIPV   10.0f
#define NEGINF  -1.0e30f

// D = A(16x4, f32) * B(4x16, f32) + C(16x16, f32)  -> v_wmma_f32_16x16x4_f32
__device__ __forceinline__ v8f wmma4(v2f a, v2f b, v8f c) {
  return __builtin_amdgcn_wmma_f32_16x16x4_f32(false, a, false, b, (short)0, c,
                                               false, false);
}

// ---------------------------------------------------------------------------
// Kernel 1: q = q_first + reshape(ELN @ Wq^T)   ->  qws [B,H,PPAD,16]
// grid (B, PT), block 256 (8 waves; wave w == head h == 16-col slice)
// ---------------------------------------------------------------------------
__global__ void __launch_bounds__(256) qlast_kernel(
    const float* __restrict__ eln, const float* __restrict__ qfirst,
    const float* __restrict__ wq, float* __restrict__ qws) {
  __shared__ float As[16 * 132];
  const int b = blockIdx.x, pt = blockIdx.y;
  const int tid = threadIdx.x;

  for (int i = tid; i < 16 * 128; i += 256) {
    int row = i >> 7, col = i & 127;
    int p = pt * 16 + row;
    As[row * 132 + col] = (p < PP) ? eln[((size_t)b * PP + p) * DD + col] : 0.f;
  }
  __syncthreads();

  const int lane = tid & 31, w = tid >> 5;
  const int kh = lane >> 4, lm = lane & 15, koff = kh * 2;

  v8f c = {0.f, 0.f, 0.f, 0.f, 0.f, 0.f, 0.f, 0.f};
#pragma unroll
  for (int kk = 0; kk < 32; ++kk) {
    int k = kk * 4 + koff;
    v2f a;  a.x = As[lm * 132 + k];  a.y = As[lm * 132 + k + 1];
    const float* wp = wq + ((size_t)(w * 16 + lm)) * DD + k;   // Wq[j][k]
    v2f bv; bv.x = wp[0]; bv.y = wp[1];
    c = wmma4(a, bv, c);
  }
#pragma unroll
  for (int r = 0; r < 8; ++r) {
    int m = r + 8 * kh;
    int p = pt * 16 + m;
    float qv = 0.f;
    if (p < PP)
      qv = c[r] + qfirst[(((size_t)b * HH + w) * PP + p) * QDD + lm];
    qws[(((size_t)b * HH + w) * PPAD + p) * QDD + lm] = qv;
  }
}

// ---------------------------------------------------------------------------
// Kernel 2: attention per (b,h,ptile). grid (PT,H,B), block 128 (4 waves).
// dyn LDS: S[16][SROW] + rowsum[16] + opart[4][16][16]
// ---------------------------------------------------------------------------
__global__ void __launch_bounds__(128) attn_kernel(
    const float* __restrict__ qws, const float* __restrict__ kmat,
    const float* __restrict__ vmat, const float* __restrict__ ninf,
    float* __restrict__ ocws) {
  extern __shared__ float sm[];
  float* S      = sm;                  // 16 x SROW
  float* rowsum = sm + 16 * SROW;      // 16
  float* opart  = rowsum + 16;         // 4*16*16 (also reduction scratch)

  const int pt = blockIdx.x, h = blockIdx.y, b = blockIdx.z;
  const int tid = threadIdx.x, lane = tid & 31, w = tid >> 5;
  const int kh = lane >> 4, lm = lane & 15, koff = kh * 2;
  const size_t bh = (size_t)b * HH + h;

  // A-operand fragments of the Q tile: A[m=lm][k], k = 4*kk + koff
  v2f qa[4];
  {
    const float* qp = qws + (bh * PPAD + (pt * 16 + lm)) * QDD + koff;
#pragma unroll
    for (int kk = 0; kk < 4; ++kk) { qa[kk].x = qp[4 * kk]; qa[kk].y = qp[4 * kk + 1]; }
  }

  // ---- phase 1: S = Q K^T / 4 + mask_full ----
  for (int nt = w; nt < NT_K; nt += 4) {
    v8f c = {0.f, 0.f, 0.f, 0.f, 0.f, 0.f, 0.f, 0.f};
#pragma unroll
    for (int kk = 0; kk < 4; ++kk) {
      int krow = nt * 16 + lm;                 // B col n=lm -> K row
      int kq = kk * 4 + koff;
      v2f bv;
      if (krow < NTOTAL) {
        const float* kp = kmat + (bh * NTOTAL + krow) * QDD + kq;
        bv.x = kp[0]; bv.y = kp[1];
      } else { bv.x = 0.f; bv.y = 0.f; }
      c = wmma4(qa[kk], bv, c);
    }
#pragma unroll
    for (int r = 0; r < 8; ++r) {
      int m = r + 8 * kh;
      int p = pt * 16 + m;
      int col = nt * 16 + lm;
      float s;
      if (col >= NTOTAL) s = 0.f;              // pad col -> final weight 0
      else {
        s = c[r] * 0.25f;                      // 1/sqrt(16)
        if (col < NN && p < PP)
          s += ninf[((size_t)b * PP + p) * NN + col];  // patch cols: mask 0
      }
      S[m * SROW + nt * 16 + lm] = s;
    }
  }
  __syncthreads();

  // ---- phase 2: softmax over cols [0,1196), 16 rows x 8 threads ----
  {
    int row = tid >> 3, t = tid & 7;
    float lmax = NEGINF;
    for (int cI = t; cI < NTOTAL; cI += 8) lmax = fmaxf(lmax, S[row * SROW + cI]);
    opart[tid] = lmax;
    __syncthreads();
    float rmax = NEGINF;
#pragma unroll
    for (int i = 0; i < 8; ++i) rmax = fmaxf(rmax, opart[(row << 3) + i]);
    __syncthreads();
    float lsum = 0.f;
    for (int cI = t; cI < NTOTAL; cI += 8) {
      float wv = __expf(S[row * SROW + cI] - rmax);
      S[row * SROW + cI] = wv;
      lsum += wv;
    }
    opart[tid] = lsum;
    __syncthreads();
    if (t == 0) {
      float rs = 0.f;
#pragma unroll
      for (int i = 0; i < 8; ++i) rs += opart[(row << 3) + i];
      rowsum[row] = rs;
    }
  }
  __syncthreads();

  // ---- phase 3: O += W V ----
  v8f o = {0.f, 0.f, 0.f, 0.f, 0.f, 0.f, 0.f, 0.f};
  for (int nt = w; nt < NT_K; nt += 4) {
#pragma unroll
    for (int kk = 0; kk < 4; ++kk) {
      int kq = kk * 4 + koff;
      v2f a, bv;
      a.x = S[lm * SROW + nt * 16 + kq];
      a.y = S[lm * SROW + nt * 16 + kq + 1];
      int vr0 = nt * 16 + kq, vr1 = vr0 + 1;
      bv.x = (vr0 < NTOTAL) ? vmat[(bh * NTOTAL + vr0) * QDD + lm] : 0.f;
      bv.y = (vr1 < NTOTAL) ? vmat[(bh * NTOTAL + vr1) * QDD + lm] : 0.f;
      o = wmma4(a, bv, o);
    }
  }
#pragma unroll
  for (int r = 0; r < 8; ++r) {
    int m = r + 8 * kh;
    opart[(w * 16 + m) * 16 + lm] = o[r];
  }
  __syncthreads();

  for (int i = tid; i < 256; i += 128) {
    int m = i >> 4, d = i & 15;
    float val = opart[m * 16 + d] + opart[(16 + m) * 16 + d] +
                opart[(32 + m) * 16 + d] + opart[(48 + m) * 16 + d];
    val /= rowsum[m];
    int p = pt * 16 + m;
    ocws[((size_t)b * PPAD + p) * DD + h * QDD + d] = (p < PP) ? val : 0.f;
  }
}

// ---------------------------------------------------------------------------
// Kernel 3: MH = OC @ mh_combine^T; S2 = MH @ SHK; logits = 10*tanh(S2/sqrt(D))
//           + edge_bias[cur] + mask; softmax over N -> probs
// grid (B, PT), block 256 (8 waves).
// ---------------------------------------------------------------------------
__global__ void __launch_bounds__(256) final_kernel(
    const float* __restrict__ ocws, const float* __restrict__ mhc,
    const float* __restrict__ shk, const float* __restrict__ edge,
    const int* __restrict__ cur, const float* __restrict__ ninf,
    float* __restrict__ out) {
  extern __shared__ float sm[];
  float* OC  = sm;                 // 16*132
  float* MH  = OC + 16 * 132;      // 16*132
  float* SC  = MH + 16 * 132;      // 16*SCROW
  float* red = SC + 16 * SCROW;    // 256

  const int b = blockIdx.x, pt = blockIdx.y;
  const int tid = threadIdx.x, lane = tid & 31, w = tid >> 5;
  const int kh = lane >> 4, lm = lane & 15, koff = kh * 2;

  for (int i = tid; i < 16 * 128; i += 256) {
    int row = i >> 7, col = i & 127;
    OC[row * 132 + col] = ocws[((size_t)b * PPAD + pt * 16 + row) * DD + col];
  }
  __syncthreads();

  // combine GEMM: wave w produces MH cols [16w, 16w+16)
  {
    v8f c = {0.f, 0.f, 0.f, 0.f, 0.f, 0.f, 0.f, 0.f};
#pragma unroll
    for (int kk = 0; kk < 32; ++kk) {
      int k = kk * 4 + koff;
      v2f a;  a.x = OC[lm * 132 + k];  a.y = OC[lm * 132 + k + 1];
      const float* mp = mhc + ((size_t)(w * 16 + lm)) * DD + k;  // mh_combine[j][k]
      v2f bv; bv.x = mp[0]; bv.y = mp[1];
      c = wmma4(a, bv, c);
    }
#pragma unroll
    for (int r = 0; r < 8; ++r)
      MH[(r + 8 * kh) * 132 + w * 16 + lm] = c[r];
  }
  __syncthreads();

  // pointer GEMM + tanh clip + edge-bias gather + mask
  for (int nt = w; nt < NT_N; nt += 8) {
    int col = nt * 16 + lm;
    bool cok = col < NN;
    v8f c = {0.f, 0.f, 0.f, 0.f, 0.f, 0.f, 0.f, 0.f};
#pragma unroll
    for (int kk = 0; kk < 32; ++kk) {
      int k = kk * 4 + koff;
      v2f a;  a.x = MH[lm * 132 + k];  a.y = MH[lm * 132 + k + 1];
      v2f bv;
      bv.x = cok ? shk[((size_t)b * DD + k) * NN + col] : 0.f;     // SHK[b][k][n]
      bv.y = cok ? shk[((size_t)b * DD + k + 1) * NN + col] : 0.f;
      c = wmma4(a, bv, c);
    }
#pragma unroll
    for (int r = 0; r < 8; ++r) {
      int m = r + 8 * kh;
      int p = pt * 16 + m;
      float s = NEGINF;
      if (cok && p < PP) {
        int node = cur[b * PP + p];
        s = CLIPV * tanhf(c[r] * INV_SQRT_D)
          + edge[((size_t)b * NN + node) * NN + col]
          + ninf[((size_t)b * PP + p) * NN + col];
      }
      SC[m * SCROW + col] = s;
    }
  }
  __syncthreads();

  // softmax over N=1000: 16 rows x 16 threads
  {
    int row = tid >> 4, t = tid & 15;
    int p = pt * 16 + row;
    bool act = p < PP;
    float lmax = NEGINF;
    if (act)
      for (int cI = t; cI < NN; cI += 16) lmax = fmaxf(lmax, SC[row * SCROW + cI]);
    red[tid] = lmax;
    __syncthreads();
    float rmax = NEGINF;
#pragma unroll
    for (int i = 0; i < 16; ++i) rmax = fmaxf(rmax, red[(row << 4) + i]);
    __syncthreads();
    float lsum = 0.f;
    if (act)
      for (int cI = t; cI < NN; cI += 16) {
        float wv = __expf(SC[row * SCROW + cI] - rmax);
        SC[row * SCROW + cI] = wv;
        lsum += wv;
      }
    red[tid] = lsum;
    __syncthreads();
    float rsum = 0.f;
#pragma unroll
    for (int i = 0; i < 16; ++i) rsum += red[(row << 4) + i];
    if (act) {
      float inv = 1.0f / rsum;
      for (int cI = t; cI < NN; cI += 16)
        out[((size_t)b * PP + p) * NN + cI] = SC[row * SCROW + cI] * inv;
    }
  }
}

// ---------------------------------------------------------------------------
extern "C" void kernel_launch(void* const* d_in, const int* in_sizes, int n_in,
                              void* d_out, int out_size, void* d_ws, size_t ws_size,
                              hipStream_t stream) {
  (void)in_sizes; (void)n_in; (void)out_size; (void)ws_size;
  const float* eln    = (const float*)d_in[0];
  const float* ninf   = (const float*)d_in[1];
  const float* qfirst = (const float*)d_in[2];
  const float* kmat   = (const float*)d_in[3];
  const float* vmat   = (const float*)d_in[4];
  const float* shk    = (const float*)d_in[5];
  const float* wq     = (const float*)d_in[6];
  const float* mhc    = (const float*)d_in[7];
  const float* edge   = (const float*)d_in[8];
  const int*   cur    = (const int*)d_in[9];

  float* qws  = (float*)d_ws;                               // B*H*PPAD*16 floats
  float* ocws = qws + (size_t)BB * HH * PPAD * QDD;         // B*PPAD*128 floats
  float* out  = (float*)d_out;

  dim3 gA(BB, PT);
  hipLaunchKernelGGL(qlast_kernel, gA, dim3(256), 0, stream, eln, qfirst, wq, qws);

  dim3 gB(PT, HH, BB);
  size_t ldsB = (size_t)(16 * SROW + 16 + 4 * 256) * sizeof(float);
  hipLaunchKernelGGL(attn_kernel, gB, dim3(128), ldsB, stream,
                     qws, kmat, vmat, ninf, ocws);

  dim3 gC(BB, PT);
  size_t ldsC = (size_t)(16 * 132 * 2 + 16 * SCROW + 256) * sizeof(float);
  hipLaunchKernelGGL(final_kernel, gC, dim3(256), ldsC, stream,
                     ocws, mhc, shk, edge, cur, ninf, out);
}